// ModBlock_51256139710781
// MI455X (gfx1250) — compile-verified
//
#include <hip/hip_runtime.h>

typedef __attribute__((ext_vector_type(2))) float v2f;
typedef __attribute__((ext_vector_type(8))) float v8f;

#define BSZ 4
#define TSZ 1024
#define DSZ 256
#define FSZ 128
#define CAT 129
#define LDH 130   // padded h1 row stride (even -> 8B-aligned float2, bank-spread)

__launch_bounds__(256, 1)
__global__ void modblock_kernel(const float* __restrict__ input,
                                const float* __restrict__ feature,
                                const float* __restrict__ W1,
                                const float* __restrict__ b1,
                                const float* __restrict__ a1p,
                                const float* __restrict__ W2,
                                const float* __restrict__ b2,
                                const float* __restrict__ a2p,
                                const float* __restrict__ Wp,
                                const float* __restrict__ bp,
                                float* __restrict__ out) {
  __shared__ float h1[DSZ * LDH];     // 133,120 B; reused for h2 per wave-strip
  __shared__ float feat_s[FSZ];
  __shared__ float base1_s[FSZ];
  __shared__ float c0_s[FSZ];
  __shared__ float b2_s[FSZ];
  __shared__ float wp0_s[CAT];
  __shared__ float wp1_s[CAT];

  const int bt   = blockIdx.x;
  const int tid  = threadIdx.x;
  const int lane = tid & 31;
  const int wv   = tid >> 5;

  const float a1    = a1p[0];
  const float a2    = a2p[0];
  const float inp_d = input[bt * DSZ + tid];   // one d per thread

  // ---- Stage 0: stage small tensors into LDS ----
  if (tid < FSZ) feat_s[tid] = feature[bt * FSZ + tid];
  if (tid < CAT) { wp0_s[tid] = Wp[tid]; wp1_s[tid] = Wp[CAT + tid]; }
  __syncthreads();

  // ---- Stage 1: base1[f] = b1[f] + W1[f,1:] . feat  (once per (b,t)) ----
  if (tid < FSZ) {
    const float* w1row = W1 + tid * CAT;
    float s = b1[tid];
#pragma unroll 8
    for (int k = 0; k < FSZ; ++k) s = fmaf(w1row[1 + k], feat_s[k], s);
    base1_s[tid] = s;
    c0_s[tid]    = w1row[0];
    b2_s[tid]    = b2[tid];
  }
  __syncthreads();

  // ---- Stage 2: h1[d,f] = prelu(c0[f]*inp[d] + base1[f], a1) into LDS ----
  {
    float* row = &h1[tid * LDH];
#pragma unroll 8
    for (int f = 0; f < FSZ; f += 2) {
      float v0 = fmaf(c0_s[f],     inp_d, base1_s[f]);
      float v1 = fmaf(c0_s[f + 1], inp_d, base1_s[f + 1]);
      v0 = (v0 >= 0.0f) ? v0 : a1 * v0;
      v1 = (v1 >= 0.0f) ? v1 : a1 * v1;
      v2f pr; pr.x = v0; pr.y = v1;
      *(v2f*)(row + f) = pr;
    }
  }
  __syncthreads();

  // ---- Stage 3: GEMM h2 = h1 @ W2^T via V_WMMA_F32_16X16X4_F32 ----
  // Wave wv owns d rows [32*wv, 32*wv+32): two 16-row tiles x eight 16-col tiles.
  const int dbase = wv * 32;
  const int lrow  = lane & 15;
  const int khalf = (lane >> 4) * 2;       // lanes 0-15 -> K{0,1}; 16-31 -> K{2,3}

  const v8f vzero = {0.f, 0.f, 0.f, 0.f, 0.f, 0.f, 0.f, 0.f};
  v8f acc[2][8];
#pragma unroll
  for (int ti = 0; ti < 2; ++ti)
#pragma unroll
    for (int gt = 0; gt < 8; ++gt) acc[ti][gt] = vzero;

  const float* r0     = &h1[(dbase + lrow) * LDH + khalf];
  const float* r1     = &h1[(dbase + 16 + lrow) * LDH + khalf];
  const float* w2base = W2 + lrow * FSZ + khalf;   // B[k,n] = W2[n,k]; row n = gt*16+lrow

#pragma unroll 2
  for (int kt = 0; kt < 32; ++kt) {
    const v2f af0 = *(const v2f*)(r0 + kt * 4);
    const v2f af1 = *(const v2f*)(r1 + kt * 4);
#pragma unroll
    for (int gt = 0; gt < 8; ++gt) {
      const v2f bf = *(const v2f*)(w2base + gt * 16 * FSZ + kt * 4);
      acc[0][gt] = __builtin_amdgcn_wmma_f32_16x16x4_f32(
          false, af0, false, bf, (short)0, acc[0][gt], false, false);
      acc[1][gt] = __builtin_amdgcn_wmma_f32_16x16x4_f32(
          false, af1, false, bf, (short)0, acc[1][gt], false, false);
    }
  }

  // ---- Stage 4: h2 = prelu(acc + b2, a2); overwrite wave-private h1 strip ----
  // C/D tile layout: VGPR v <-> M = v (+8 for lanes 16-31), lane%16 <-> N.
  const int mofs = (lane >> 4) * 8;
#pragma unroll
  for (int ti = 0; ti < 2; ++ti) {
#pragma unroll
    for (int gt = 0; gt < 8; ++gt) {
      const int   g  = gt * 16 + lrow;
      const float bb = b2_s[g];
#pragma unroll
      for (int v = 0; v < 8; ++v) {
        const int d = dbase + ti * 16 + mofs + v;
        float val = acc[ti][gt][v] + bb;
        val = (val >= 0.0f) ? val : a2 * val;
        h1[d * LDH + g] = val;
      }
    }
  }
  __syncthreads();

  // ---- Stage 5: p = Wp @ [h2; inp] + bp; gate; residual ----
  {
    const float* row = &h1[tid * LDH];
    float p0 = fmaf(wp0_s[FSZ], inp_d, bp[0]);
    float p1 = fmaf(wp1_s[FSZ], inp_d, bp[1]);
#pragma unroll 8
    for (int f = 0; f < FSZ; f += 2) {
      const v2f h = *(const v2f*)(row + f);
      p0 = fmaf(wp0_s[f], h.x, fmaf(wp0_s[f + 1], h.y, p0));
      p1 = fmaf(wp1_s[f], h.x, fmaf(wp1_s[f + 1], h.y, p1));
    }
    const float sig  = 1.0f / (1.0f + expf(-p1));
    const float gate = p0 * sig;
    out[bt * DSZ + tid] = inp_d * (1.0f + gate);
  }
}

extern "C" void kernel_launch(void* const* d_in, const int* in_sizes, int n_in,
                              void* d_out, int out_size, void* d_ws, size_t ws_size,
                              hipStream_t stream) {
  const float* input   = (const float*)d_in[0];
  const float* feature = (const float*)d_in[1];
  // d_in[2] = slider (unused by the reference computation)
  const float* W1 = (const float*)d_in[3];
  const float* b1 = (const float*)d_in[4];
  const float* a1 = (const float*)d_in[5];
  const float* W2 = (const float*)d_in[6];
  const float* b2 = (const float*)d_in[7];
  const float* a2 = (const float*)d_in[8];
  const float* Wp = (const float*)d_in[9];
  const float* bp = (const float*)d_in[10];
  float* out = (float*)d_out;

  dim3 grid(BSZ * TSZ);   // one workgroup per (b,t)
  dim3 block(256);
  modblock_kernel<<<grid, block, 0, stream>>>(input, feature, W1, b1, a1,
                                              W2, b2, a2, Wp, bp, out);
}